// RGCN_63617055588531
// MI455X (gfx1250) — compile-verified
//
#include <hip/hip_runtime.h>

#define N_NODES 100000
#define DIM     128
#define NREL    8
#define MT      5            // node tiles (16 rows each) per wave
#define ROWS    (MT * 16)    // 80 nodes per block; 100000 = 80 * 1250 exactly

typedef __attribute__((ext_vector_type(2))) float v2f;
typedef __attribute__((ext_vector_type(8))) float v8f;

// Per block column r:
//   r in [0,8):  xt[r] = x @ W[r]     (written to xt buffer)
//   r == 8:      agg   = x @ SW + b   (self-loop init of accumulator)
// Block tile: 80 nodes x 128 cols. 8 waves; wave w owns cols [16w,16w+16) and
// all 5 row tiles. B fragments (32 k-steps) are held in registers for the whole
// block (64 VGPRs), so each k-step is 5 ds_load_b64 + 5 v_wmma_f32_16x16x4_f32.
__global__ __launch_bounds__(256) void rgcn_gemm(const float* __restrict__ x,
                                                 const float* __restrict__ w,    // [8,128,128]
                                                 const float* __restrict__ sw,   // [128,128]
                                                 const float* __restrict__ bias, // [128]
                                                 float* __restrict__ xt,         // [8,N,128]
                                                 float* __restrict__ agg)        // [N,128]
{
    __shared__ float As[ROWS][132];  // stride 132 -> conflict-free column reads

    const int r     = blockIdx.y;
    const int node0 = blockIdx.x * ROWS;
    const int t     = threadIdx.x;

    // Stage A tile: 80x128 floats = 2560 float4, 10 per thread, fully coalesced.
    const float4* x4 = (const float4*)(x + (size_t)node0 * DIM);
#pragma unroll
    for (int i = 0; i < 10; ++i) {
        const int idx = t + 256 * i;
        const int row = idx >> 5;          // 32 float4 per row
        const int c4  = idx & 31;
        *(float4*)&As[row][c4 * 4] = x4[row * 32 + c4];
    }
    __syncthreads();

    const int wave  = t >> 5;
    const int lane  = t & 31;
    const int col0  = wave * 16;
    const int rowA  = lane & 15;
    const int khalf = (lane >> 4) * 2;     // lanes 16-31 hold the K+2,K+3 halves
    const int col   = col0 + rowA;

    const float* wbase = (r < NREL) ? (w + (size_t)r * DIM * DIM) : sw;

    // Preload all B fragments (4x16 f32 each) for this wave's 16 columns.
    v2f bf[32];
#pragma unroll
    for (int ks = 0; ks < 32; ++ks) {
        const int kk = ks * 4 + khalf;
        bf[ks].x = wbase[kk * DIM + col];
        bf[ks].y = wbase[(kk + 1) * DIM + col];
    }

    v8f acc[MT];
#pragma unroll
    for (int m = 0; m < MT; ++m) acc[m] = (v8f){};

#pragma unroll
    for (int ks = 0; ks < 32; ++ks) {
        const int kk = ks * 4 + khalf;
#pragma unroll
        for (int m = 0; m < MT; ++m) {
            // A frag (16x4 f32): lane<16 -> {A[m][k],A[m][k+1]}, lane>=16 -> {+2,+3}
            const float2 av = *(const float2*)&As[rowA + 16 * m][kk];
            v2f a;
            a.x = av.x;
            a.y = av.y;
            acc[m] = __builtin_amdgcn_wmma_f32_16x16x4_f32(false, a, false, bf[ks],
                                                           (short)0, acc[m], false, false);
        }
    }

    // C/D layout: VGPR v, lanes 0-15 -> row v, lanes 16-31 -> row v+8; col = lane%16.
    const int rowHi = (lane >> 4) * 8;
    if (r < NREL) {
        float* outp = xt + ((size_t)r * N_NODES + node0) * DIM;
#pragma unroll
        for (int m = 0; m < MT; ++m)
#pragma unroll
            for (int v = 0; v < 8; ++v)
                outp[(size_t)(m * 16 + v + rowHi) * DIM + col] = acc[m][v];
    } else {
        float* outp = agg + (size_t)node0 * DIM;
        const float bb = bias[col];
#pragma unroll
        for (int m = 0; m < MT; ++m)
#pragma unroll
            for (int v = 0; v < 8; ++v)
                outp[(size_t)(m * 16 + v + rowHi) * DIM + col] = acc[m][v] + bb;
    }
}

// One wave per edge: agg[dst] += xt[rel, src]. 512B coalesced gather per edge,
// 128 hardware f32 atomics that resolve in L2 (agg table = 51MB, L2-resident).
__global__ __launch_bounds__(256) void rgcn_scatter(const float* __restrict__ xt,
                                                    const int* __restrict__ src,
                                                    const int* __restrict__ dst,
                                                    const int* __restrict__ rel,
                                                    float* __restrict__ agg,
                                                    int E)
{
    const int wave = threadIdx.x >> 5;
    const int lane = threadIdx.x & 31;
    const int e    = blockIdx.x * 8 + wave;
    if (e >= E) return;

    const int rr = rel[e];
    const int s  = src[e];
    const int d  = dst[e];

    const float4 m = *(const float4*)(xt + ((size_t)rr * N_NODES + s) * DIM + lane * 4);
    float* o = agg + (size_t)d * DIM + lane * 4;
    unsafeAtomicAdd(o + 0, m.x);
    unsafeAtomicAdd(o + 1, m.y);
    unsafeAtomicAdd(o + 2, m.z);
    unsafeAtomicAdd(o + 3, m.w);
}

__global__ __launch_bounds__(256) void rgcn_relu(const float* __restrict__ in,
                                                 float* __restrict__ out, int n4)
{
    const int i = blockIdx.x * 256 + threadIdx.x;
    if (i < n4) {
        float4 v = ((const float4*)in)[i];
        v.x = fmaxf(v.x, 0.0f);
        v.y = fmaxf(v.y, 0.0f);
        v.z = fmaxf(v.z, 0.0f);
        v.w = fmaxf(v.w, 0.0f);
        ((float4*)out)[i] = v;
    }
}

extern "C" void kernel_launch(void* const* d_in, const int* in_sizes, int n_in,
                              void* d_out, int out_size, void* d_ws, size_t ws_size,
                              hipStream_t stream) {
    const float* emb = (const float*)d_in[0];   // [N,128]
    const float* w1  = (const float*)d_in[1];   // [8,128,128]
    const float* sl1 = (const float*)d_in[2];   // [128,128]
    const float* b1  = (const float*)d_in[3];   // [128]
    const float* w2  = (const float*)d_in[4];
    const float* sl2 = (const float*)d_in[5];
    const float* b2  = (const float*)d_in[6];
    const int*   edge_index = (const int*)d_in[7];  // [2,E] flat
    const int*   edge_type  = (const int*)d_in[8];  // [E]
    const int E = in_sizes[8];

    float* ws  = (float*)d_ws;
    float* xt  = ws;                                      // 8*N*128 floats (410 MB)
    float* agg = ws + (size_t)NREL * N_NODES * DIM;       // N*128 floats
    float* x1  = agg + (size_t)N_NODES * DIM;             // N*128 floats
    float* out = (float*)d_out;

    const dim3 ggrid(N_NODES / ROWS, NREL + 1);           // 1250 x 9 blocks
    const int  sblocks = (E + 7) / 8;                     // 8 edges (waves) per block
    const int  n4      = N_NODES * DIM / 4;
    const int  rblocks = (n4 + 255) / 256;

    const int* esrc = edge_index;
    const int* edst = edge_index + E;

    // Layer 1
    rgcn_gemm<<<ggrid, 256, 0, stream>>>(emb, w1, sl1, b1, xt, agg);
    rgcn_scatter<<<sblocks, 256, 0, stream>>>(xt, esrc, edst, edge_type, agg, E);
    rgcn_relu<<<rblocks, 256, 0, stream>>>(agg, x1, n4);

    // Layer 2
    rgcn_gemm<<<ggrid, 256, 0, stream>>>(x1, w2, sl2, b2, xt, agg);
    rgcn_scatter<<<sblocks, 256, 0, stream>>>(xt, esrc, edst, edge_type, agg, E);
    rgcn_relu<<<rblocks, 256, 0, stream>>>(agg, out, n4);
}